// GCN_18124761989811
// MI455X (gfx1250) — compile-verified
//
#include <hip/hip_runtime.h>
#include <math.h>

typedef __attribute__((ext_vector_type(2))) float v2f;
typedef __attribute__((ext_vector_type(8))) float v8f;

__device__ __forceinline__ void atomic_add_f32(float* p, float v) {
    __hip_atomic_fetch_add(p, v, __ATOMIC_RELAXED, __HIP_MEMORY_SCOPE_AGENT);
}

// ---------------- degree / normalization ----------------

__global__ void gcn_deg_init(float* __restrict__ deg, int n) {
    int i = blockIdx.x * blockDim.x + threadIdx.x;
    if (i < n) deg[i] = 1.0f;  // self-loop weight
}

__global__ void gcn_deg_edges(const int* __restrict__ col, const float* __restrict__ ew,
                              float* __restrict__ deg, int e) {
    int i = blockIdx.x * blockDim.x + threadIdx.x;
    if (i < e) atomic_add_f32(&deg[col[i]], ew[i]);
}

__global__ void gcn_dinv(const float* __restrict__ deg, float* __restrict__ dinv, int n) {
    int i = blockIdx.x * blockDim.x + threadIdx.x;
    if (i < n) {
        float d = deg[i];
        dinv[i] = (d > 0.0f) ? rsqrtf(d) : 0.0f;
    }
}

// ---------------- dense transform: out = A[n,64] @ W[64,OUT_COLS] ----------------
// MODE 0: plain store; MODE 1: +bias, relu; MODE 2: +bias, sigmoid.
// One wave32 computes one 16x16 tile via 16 chained V_WMMA_F32_16X16X4_F32.
// Requires n % 16 == 0 (N_NODES = 50000 = 3125 * 16).

template <int OUT_COLS, int MODE>
__global__ void gcn_gemm_wmma(const float* __restrict__ A, const float* __restrict__ W,
                              const float* __restrict__ bias, float* __restrict__ out,
                              int numTiles) {
    __shared__ float Wl[64 * OUT_COLS];
    for (int idx = threadIdx.x; idx < 64 * OUT_COLS; idx += blockDim.x)
        Wl[idx] = W[idx];
    __syncthreads();

    const int wave = threadIdx.x >> 5;
    const int lane = threadIdx.x & 31;
    const int half = lane >> 4;   // 0: lanes 0-15, 1: lanes 16-31
    const int ln   = lane & 15;

    int tile = blockIdx.x * (blockDim.x >> 5) + wave;
    if (tile >= numTiles) return;  // wave-uniform: EXEC stays all-1s for WMMA

    constexpr int CT = OUT_COLS / 16;
    const int rowBase = (tile / CT) * 16;
    const int colBase = (tile % CT) * 16;

    // A fragment source: row m = ln, K pair selected by lane half (ISA 16x4 f32 layout)
    const float* arow = A + (size_t)(rowBase + ln) * 64;

    v8f acc = {};
#pragma unroll
    for (int k0 = 0; k0 < 64; k0 += 4) {
        const int kb = k0 + half * 2;
        v2f a, b;
        a.x = arow[kb];
        a.y = arow[kb + 1];
        // B fragment (4x16): row K across VGPRs / lane halves, col N across lanes
        b.x = Wl[kb * OUT_COLS + colBase + ln];
        b.y = Wl[(kb + 1) * OUT_COLS + colBase + ln];
        acc = __builtin_amdgcn_wmma_f32_16x16x4_f32(
            /*neg_a=*/false, a, /*neg_b=*/false, b,
            /*c_mod=*/(short)0, acc, /*reuse_a=*/false, /*reuse_b=*/false);
    }

    float bv = 0.0f;
    if constexpr (MODE != 0) bv = bias[colBase + ln];

#pragma unroll
    for (int v = 0; v < 8; ++v) {
        // C/D layout: VGPR v holds M = v + 8*half, N = ln
        const int m = rowBase + v + half * 8;
        float val = acc[v];
        if constexpr (MODE != 0) val += bv;
        if constexpr (MODE == 1) val = fmaxf(val, 0.0f);
        if constexpr (MODE == 2) val = 1.0f / (1.0f + expf(-val));
        out[(size_t)m * OUT_COLS + colBase + ln] = val;
    }
}

// ---------------- aggregation ----------------

// acc[i,:] = dinv[i]^2 * hW[i,:]   (self-loop term, w = 1)
__global__ void gcn_agg_init(const float* __restrict__ hW, const float* __restrict__ dinv,
                             float* __restrict__ acc, int nf) {
    int idx = blockIdx.x * blockDim.x + threadIdx.x;
    if (idx < nf) {
        const int i = idx >> 6;
        const float di = dinv[i];
        acc[idx] = di * di * hW[idx];
    }
}

// One wave per edge: norm = dinv[r]*w*dinv[c]; acc[c,:] += norm * hW[r,:]
__global__ void gcn_agg_edges(const int* __restrict__ row, const int* __restrict__ col,
                              const float* __restrict__ ew, const float* __restrict__ dinv,
                              const float* __restrict__ hW, float* __restrict__ acc, int e) {
    const int eid  = blockIdx.x * (blockDim.x >> 5) + (threadIdx.x >> 5);
    const int lane = threadIdx.x & 31;
    if (eid >= e) return;
    const int r = row[eid];
    const int c = col[eid];
    const float nrm = dinv[r] * ew[eid] * dinv[c];
    const float* __restrict__ src = hW + (size_t)r * 64;
    float* __restrict__ dst = acc + (size_t)c * 64;
    atomic_add_f32(&dst[lane],      nrm * src[lane]);
    atomic_add_f32(&dst[lane + 32], nrm * src[lane + 32]);
}

// out = relu(acc + b)
__global__ void gcn_bias_relu(const float* __restrict__ acc, const float* __restrict__ b,
                              float* __restrict__ out, int nf) {
    int idx = blockIdx.x * blockDim.x + threadIdx.x;
    if (idx < nf) out[idx] = fmaxf(acc[idx] + b[idx & 63], 0.0f);
}

// ---------------- launch ----------------

extern "C" void kernel_launch(void* const* d_in, const int* in_sizes, int n_in,
                              void* d_out, int out_size, void* d_ws, size_t ws_size,
                              hipStream_t stream) {
    (void)n_in; (void)out_size; (void)ws_size;

    const float* x   = (const float*)d_in[0];
    const int*   ei  = (const int*)d_in[1];
    const float* ew  = (const float*)d_in[2];
    const float* W1  = (const float*)d_in[3];
    const float* b1  = (const float*)d_in[4];
    const float* W2  = (const float*)d_in[5];
    const float* b2  = (const float*)d_in[6];
    const float* Wm1 = (const float*)d_in[7];
    const float* bm1 = (const float*)d_in[8];
    const float* Wm2 = (const float*)d_in[9];
    const float* bm2 = (const float*)d_in[10];
    float* out = (float*)d_out;

    const int n = in_sizes[0] / 64;   // 50000
    const int e = in_sizes[1] / 2;    // 800000
    const int* row = ei;
    const int* col = ei + e;

    float* ws   = (float*)d_ws;
    float* deg  = ws;
    float* dinv = deg + n;
    float* bufA = dinv + n;                 // hW (dense-transform result)
    float* bufB = bufA + (size_t)n * 64;    // aggregation accumulator
    float* bufC = bufB + (size_t)n * 64;    // layer output

    const int nf = n * 64;
    const int gN  = (n  + 255) / 256;
    const int gE  = (e  + 255) / 256;
    const int gNF = (nf + 255) / 256;
    const int gEw = (e + 7) / 8;            // 8 edges (waves) per 256-thread block

    const int tiles64  = (n / 16) * 4;      // 16x16 tiles for OUT_COLS=64
    const int tiles16  = (n / 16);          // for OUT_COLS=16
    const int gT64 = (tiles64 + 7) / 8;
    const int gT16 = (tiles16 + 7) / 8;

    // Degree + normalization (shared by both conv layers)
    gcn_deg_init <<<gN, 256, 0, stream>>>(deg, n);
    gcn_deg_edges<<<gE, 256, 0, stream>>>(col, ew, deg, e);
    gcn_dinv     <<<gN, 256, 0, stream>>>(deg, dinv, n);

    // Conv layer 1: h1 = relu(agg(x @ W1) + b1)
    gcn_gemm_wmma<64, 0><<<gT64, 256, 0, stream>>>(x, W1, nullptr, bufA, tiles64);
    gcn_agg_init <<<gNF, 256, 0, stream>>>(bufA, dinv, bufB, nf);
    gcn_agg_edges<<<gEw, 256, 0, stream>>>(row, col, ew, dinv, bufA, bufB, e);
    gcn_bias_relu<<<gNF, 256, 0, stream>>>(bufB, b1, bufC, nf);

    // Conv layer 2: h2 = relu(agg(h1 @ W2) + b2)
    gcn_gemm_wmma<64, 0><<<gT64, 256, 0, stream>>>(bufC, W2, nullptr, bufA, tiles64);
    gcn_agg_init <<<gNF, 256, 0, stream>>>(bufA, dinv, bufB, nf);
    gcn_agg_edges<<<gEw, 256, 0, stream>>>(row, col, ew, dinv, bufA, bufB, e);
    gcn_bias_relu<<<gNF, 256, 0, stream>>>(bufB, b2, bufC, nf);

    // MLP: h3 = relu(h2 @ Wm1 + bm1); out = sigmoid(h3 @ Wm2 + bm2)
    gcn_gemm_wmma<64, 1><<<gT64, 256, 0, stream>>>(bufC, Wm1, bm1, bufA, tiles64);
    gcn_gemm_wmma<16, 2><<<gT16, 256, 0, stream>>>(bufA, Wm2, bm2, out, tiles16);
}